// jump_loss_full_63359357551401
// MI455X (gfx1250) — compile-verified
//
#include <hip/hip_runtime.h>

// CDNA5 / gfx1250 (MI455X), wave32. V_WMMA_F32_16X16X4_F32 computes the
// batched (16 pts x 64) @ (64 x 3) GEMVs of both MLPs (forward + JVP),
// stacked along K=256 into one 16x16 f32 accumulator per 16-point tile.
// B operands live entirely in registers, gated with cndmask. Activations
// use the hardware v_tanh_f32 (TRANS pipe, co-executes with XDL WMMA);
// epilogue transcendentals are branch-free v_log/v_exp/v_rcp/v_sqrt.

typedef float v2f __attribute__((ext_vector_type(2)));
typedef float v4f __attribute__((ext_vector_type(4)));
typedef float v8f __attribute__((ext_vector_type(8)));

#define N_PTS (1 << 20)

__device__ __forceinline__ float fast_tanh(float x) {
#if __has_builtin(__builtin_amdgcn_tanhf)
  return __builtin_amdgcn_tanhf(x);            // v_tanh_f32
#else
  // branch-free: tanh(x) = sign(x) * (1 - t) / (1 + t), t = exp2(-2*log2(e)*|x|)
  const float ax = __builtin_fabsf(x);
  const float t  = __builtin_amdgcn_exp2f(ax * -2.8853900817779268f);
  const float y  = (1.0f - t) * __builtin_amdgcn_rcpf(1.0f + t);
  return __builtin_copysignf(y, x);
#endif
}

__global__ void zero_out_kernel(float* out) { out[0] = 0.0f; }

__global__ __launch_bounds__(256) void jump_loss_wmma_kernel(
    const float* __restrict__ u_y1,  const float* __restrict__ u_ym1,
    const float* __restrict__ u_x1,  const float* __restrict__ u_xm1,
    const float* __restrict__ sW1,   const float* __restrict__ sB1,
    const float* __restrict__ sW2,   const float* __restrict__ sB2,
    const float* __restrict__ jW1,   const float* __restrict__ jB1,
    const float* __restrict__ jW2,   const float* __restrict__ jB2,
    float* __restrict__ out)
{
  // Packed per-(half,j) first-layer params, 12 floats per row (48B, 16B-aligned):
  // [w1s_nz k0,k0+1 | b1s k0,k0+1 | w1s_t k0,k0+1 | w1j_nz k0,k0+1 | b1j k0,k0+1 | w1j_t k0,k0+1]
  __shared__ __align__(16) float apar[2 * 16 * 12];
  __shared__ float w2s[192], w2j[192];
  __shared__ float bias2[6];
  __shared__ float dt[8][256];   // per-wave 16x16 f32 D tile

  const int tid  = threadIdx.x;
  const int edge = blockIdx.x >> 13;          // 8192 blocks per edge
  const int inz  = edge >> 1;                 // nonzero coord: 0 jump_y, 1 jump_x
  const float sgn = (edge & 1) ? -1.0f : 1.0f;

  // ---- stage weights ----
  for (int i = tid; i < 384; i += 256) {
    const int hlf  = i / 192;
    const int rem  = i % 192;
    const int j    = rem / 12;
    const int slot = rem % 12;
    const int grp  = slot >> 1;               // 0..5
    const int k    = 4 * j + 2 * hlf + (slot & 1);
    float v;
    if      (grp == 0) v = sW1[inz * 64 + k];
    else if (grp == 1) v = sB1[k];
    else if (grp == 2) v = sW1[(1 - inz) * 64 + k];
    else if (grp == 3) v = jW1[inz * 64 + k];
    else if (grp == 4) v = jB1[k];
    else               v = jW1[(1 - inz) * 64 + k];
    apar[i] = v;
  }
  if (tid < 192) { w2s[tid] = sW2[tid]; w2j[tid] = jW2[tid]; }
  if (tid < 3)   { bias2[tid] = sB2[tid]; bias2[3 + tid] = jB2[tid]; }
  __syncthreads();

  const float* uptr = (edge == 0) ? u_y1 : (edge == 1) ? u_ym1
                    : (edge == 2) ? u_x1 : u_xm1;

  const int wv   = tid >> 5;        // wave in block (8 waves)
  const int lane = tid & 31;
  const int half = lane >> 4;       // 0: lanes 0-15, 1: lanes 16-31
  const int col  = lane & 15;

  const int tile = (blockIdx.x & 8191) * 8 + wv;   // tile of 16 points
  const int m0   = tile << 4;

  const float u = uptr[m0 + col];   // this lane's point coordinate magnitude
  const float c = sgn * u;          // signed coordinate fed to the MLPs

  // ---- per-lane B operands: this column is nonzero in exactly one segment ----
  const int  sa  = col / 3;                    // active segment (>=4 -> never)
  const int  saq = (sa < 4) ? sa : 0;
  const int  ci  = (sa < 4) ? (col - 3 * sa) : 0;
  const float* W2act = (saq < 2) ? w2s : w2j;
  const int  off = half << 1;                  // K offset in each K=4 group
  float Breg[32];
#pragma unroll
  for (int j = 0; j < 16; ++j) {
    const int k0 = 4 * j + off;
    Breg[2*j]     = W2act[k0 * 3 + ci];
    Breg[2*j + 1] = W2act[(k0 + 1) * 3 + ci];
  }
  const bool ms0 = (sa == 0), ms1 = (sa == 1), ms2 = (sa == 2), ms3 = (sa == 3);

  // ---- stacked-K GEMM: D[16 pts][16 ch] = A[16x256] * B[256x16] ----
  // K segments: 0=h_sing(ch0-2) 1=g_sing(ch3-5) 2=h_jump(ch6-8) 3=g_jump(ch9-11)
  const int rowbase = half * 192;              // floats
  v8f acc = {};
#pragma unroll
  for (int j = 0; j < 16; ++j) {
    const int rb = rowbase + j * 12;
    const v4f q0 = *(const v4f*)&apar[rb + 0];   // wnzS0 wnzS1 bS0 bS1
    const v4f q1 = *(const v4f*)&apar[rb + 4];   // wtS0 wtS1 wnzJ0 wnzJ1
    const v4f q2 = *(const v4f*)&apar[rb + 8];   // bJ0 bJ1 wtJ0 wtJ1

    const float h0 = fast_tanh(fmaf(c, q0.x, q0.z));
    const float h1 = fast_tanh(fmaf(c, q0.y, q0.w));
    const float g0 = (1.0f - h0 * h0) * q1.x;
    const float g1 = (1.0f - h1 * h1) * q1.y;
    const float H0 = fast_tanh(fmaf(c, q1.z, q2.x));
    const float H1 = fast_tanh(fmaf(c, q1.w, q2.y));
    const float G0 = (1.0f - H0 * H0) * q2.z;
    const float G1 = (1.0f - H1 * H1) * q2.w;

    const float b0 = Breg[2*j], b1 = Breg[2*j + 1];

    v2f A, B;
    A = (v2f){ h0, h1 };
    B = (v2f){ ms0 ? b0 : 0.0f, ms0 ? b1 : 0.0f };
    acc = __builtin_amdgcn_wmma_f32_16x16x4_f32(false, A, false, B, (short)0, acc, false, false);
    A = (v2f){ g0, g1 };
    B = (v2f){ ms1 ? b0 : 0.0f, ms1 ? b1 : 0.0f };
    acc = __builtin_amdgcn_wmma_f32_16x16x4_f32(false, A, false, B, (short)0, acc, false, false);
    A = (v2f){ H0, H1 };
    B = (v2f){ ms2 ? b0 : 0.0f, ms2 ? b1 : 0.0f };
    acc = __builtin_amdgcn_wmma_f32_16x16x4_f32(false, A, false, B, (short)0, acc, false, false);
    A = (v2f){ G0, G1 };
    B = (v2f){ ms3 ? b0 : 0.0f, ms3 ? b1 : 0.0f };
    acc = __builtin_amdgcn_wmma_f32_16x16x4_f32(false, A, false, B, (short)0, acc, false, false);
  }

  // ---- spill D tile: (VGPR r, lane) holds D[r + 8*half][col] ----
#pragma unroll
  for (int r = 0; r < 8; ++r)
    dt[wv][(r + 8 * half) * 16 + col] = acc[r];
  __builtin_amdgcn_wave_barrier();

  float term = 0.0f;
  if (lane < 16) {
    const int m = col;
    const float os1 = dt[wv][m*16 + 1] + bias2[1];  // sing px
    const float os2 = dt[wv][m*16 + 2] + bias2[2];  // sing py
    const float ts0 = dt[wv][m*16 + 3];             // JVP sing p1
    const float ts1 = dt[wv][m*16 + 4];             // JVP sing px
    const float ts2 = dt[wv][m*16 + 5];             // JVP sing py
    const float oj1 = dt[wv][m*16 + 7] + bias2[4];  // jump px
    const float oj2 = dt[wv][m*16 + 8] + bias2[5];  // jump py
    const float tj0 = dt[wv][m*16 + 9];             // JVP jump p1
    const float tj1 = dt[wv][m*16 + 10];            // JVP jump px
    const float tj2 = dt[wv][m*16 + 11];            // JVP jump py

    // r in [1e-6, ~1.42]: strictly positive & normal -> branch-free trans ops
    const float r    = __builtin_amdgcn_sqrtf(fmaf(c, c, 1e-12f));
    const float rcpr = __builtin_amdgcn_rcpf(r);
    const float lg   = __builtin_amdgcn_logf(r);                       // log2(r)
    const float rl   = __builtin_amdgcn_exp2f(0.66666666666666663f * lg);  // r^(2/3)
    const float bcv  = 1.0f - c * c;                                   // bc on boundary
    const float rcb  = fmaxf(1.0f - 2.0f * r, 0.0f);                   // 1 - r/R0
    const float rc   = rcb * rcb;
    const float sf   = rc * __builtin_amdgcn_exp2f(-0.33333333333333331f * lg); // rc*r^(l-1)
    const float a    = u * rcpr;                                       // |c|/r

    const float tsA = (inz == 0) ? ts1 : ts2;        // tangent of lelu channel
    const float tjA = (inz == 0) ? tj1 : tj2;
    const float osC = (inz == 0) ? os2 : os1;        // corr channel (2-inz)
    const float ojC = (inz == 0) ? oj2 : oj1;

    const float dphi = (ts0 + tsA * a) * rc * rl + bcv * (tj0 + tjA * u);
    const float corr = ojC * bcv + osC * sf;
    const float dmn  = dphi - corr;                  // dleft / dbottom
    const float dpl  = dphi + corr;                  // dright / dtop
    const float omega = fminf(u * 40.0f, 1.0f);      // clamp(u/0.025, 0, 1)

    float cmu, cpu;
    if      (edge == 0) { cmu = 4.0f; cpu = 1.0f; }  // (S4, S1)
    else if (edge == 1) { cmu = 3.0f; cpu = 2.0f; }  // (S3, S2)
    else if (edge == 2) { cmu = 2.0f; cpu = 1.0f; }  // (S2, S1)
    else                { cmu = 3.0f; cpu = 4.0f; }  // (S3, S4)

    const float dd = cmu * dmn - cpu * dpl;
    term = dd * dd * omega;
  }

  // wave32 reduction, then one atomic per wave
#pragma unroll
  for (int o = 16; o > 0; o >>= 1) term += __shfl_down(term, o, 32);
  if (lane == 0) atomicAdd(out, term * (1.0f / (float)N_PTS));
}

extern "C" void kernel_launch(void* const* d_in, const int* in_sizes, int n_in,
                              void* d_out, int out_size, void* d_ws, size_t ws_size,
                              hipStream_t stream) {
  (void)in_sizes; (void)n_in; (void)out_size; (void)d_ws; (void)ws_size;
  float* out = (float*)d_out;
  zero_out_kernel<<<dim3(1), dim3(1), 0, stream>>>(out);
  // 4 edges * 8192 blocks; each block = 8 waves * 16 points = 128 points
  jump_loss_wmma_kernel<<<dim3(32768), dim3(256), 0, stream>>>(
      (const float*)d_in[0], (const float*)d_in[1],
      (const float*)d_in[2], (const float*)d_in[3],
      (const float*)d_in[4], (const float*)d_in[5],
      (const float*)d_in[6], (const float*)d_in[7],
      (const float*)d_in[8], (const float*)d_in[9],
      (const float*)d_in[10], (const float*)d_in[11],
      out);
}